// ButterflyLayer_71425306132665
// MI455X (gfx1250) — compile-verified
//
#include <hip/hip_runtime.h>

typedef float v2f __attribute__((ext_vector_type(2)));
typedef float v8f __attribute__((ext_vector_type(8)));

constexpr int NLVL = 8;
constexpr int C    = 64;
constexpr int FIN  = 16;
constexpr int FOUT = 16;
constexpr int B    = 512;
constexpr int KMAX = 256;            // 2^NLVL leaves / time slots
constexpr int ROWS = B * KMAX;       // 131072 (b,k,t)-rows at every level
constexpr int WTP  = 130;            // LDS pitch (floats) for transposed 128-K rows

__device__ __forceinline__ v8f wmma4(v2f a, v2f b, v8f c) {
  // D(16x16,f32) = A(16x4,f32) * B(4x16,f32) + C   — full fp32 matrix op
  return __builtin_amdgcn_wmma_f32_16x16x4_f32(false, a, false, b, (short)0, c,
                                               false, false);
}

// ---------------- Input layer: V0[b*256+t][c] = relu(x_row(16) @ W_in + b_in)
__global__ __launch_bounds__(256) void butterfly_input_kernel(
    const float* __restrict__ x, const float* __restrict__ Win,
    const float* __restrict__ bin, float* __restrict__ V) {
  const int tid  = threadIdx.x;
  const int wave = tid >> 5, lane = tid & 31;
  const int half = lane >> 4, l = lane & 15;
  const int r0   = (blockIdx.x * 8 + wave) * 16;         // 16-row tile base
  const float* arow = x + (size_t)(r0 + l) * FIN;

  v8f acc[4] = {};
#pragma unroll
  for (int kk = 0; kk < FIN; kk += 4) {
    const v2f a = *(const v2f*)(arow + kk + 2 * half);
#pragma unroll
    for (int nt = 0; nt < 4; ++nt) {
      v2f b;
      b.x = Win[(kk + 2 * half + 0) * C + nt * 16 + l];
      b.y = Win[(kk + 2 * half + 1) * C + nt * 16 + l];
      acc[nt] = wmma4(a, b, acc[nt]);
    }
  }
#pragma unroll
  for (int nt = 0; nt < 4; ++nt) {
    const float bias = bin[nt * 16 + l];
#pragma unroll
    for (int j = 0; j < 8; ++j) {
      const int row = r0 + j + 8 * half;
      float v = acc[nt][j] + bias;
      v = v > 0.f ? v : 0.f;
      V[(size_t)row * C + nt * 16 + l] = v;
    }
  }
}

// ---------------- One butterfly level.
// Vin layout  : [b][K/2][2T][C]  (t fastest) -> each A-row is 128 contiguous floats
// Vout layout : [b][K]  [T] [C]
// Weights W[k] (128x64) staged TRANSPOSED in LDS: Wt[d][r], pitch 130 floats,
// so a B fragment (two consecutive K values, one column) is a single ds_load_b64.
// Each wave computes an M=32 x N=64 tile: B fragments amortized over 2 row tiles.
__global__ __launch_bounds__(256) void butterfly_level_kernel(
    const float* __restrict__ Vin, float* __restrict__ Vout,
    const float* __restrict__ Wl /* pre-offset by level */,
    const float* __restrict__ bl /* pre-offset by level */, int tshift) {
  extern __shared__ float Wt[];                           // [64][WTP]
  const int tid = threadIdx.x;
  const int k   = blockIdx.y;
  const int T   = 1 << tshift;

  const float* Wg = Wl + (size_t)k * 2 * C * C;
  for (int idx = tid; idx < 2 * C * C; idx += 256) {
    const int r = idx >> 6, d = idx & 63;                 // W[r][d] -> Wt[d][r]
    Wt[d * WTP + r] = Wg[idx];
  }
  __syncthreads();

  const int wave = tid >> 5, lane = tid & 31;
  const int half = lane >> 4, l = lane & 15;
  const int r0   = (blockIdx.x * 8 + wave) * 32;          // 32-row tile base
  const int kbase = (k >> 1) << (tshift + 1);             // parent node offset

  const int rl0 = r0 + l;
  const int rl1 = r0 + 16 + l;
  const float* arow0 =
      Vin + ((size_t)(rl0 >> tshift) * 256 + kbase + 2 * (rl0 & (T - 1))) * C;
  const float* arow1 =
      Vin + ((size_t)(rl1 >> tshift) * 256 + kbase + 2 * (rl1 & (T - 1))) * C;

  v8f acc0[4] = {}, acc1[4] = {};
#pragma unroll 4
  for (int kk = 0; kk < 2 * C; kk += 4) {
    const v2f a0 = *(const v2f*)(arow0 + kk + 2 * half);
    const v2f a1 = *(const v2f*)(arow1 + kk + 2 * half);
#pragma unroll
    for (int nt = 0; nt < 4; ++nt) {
      const v2f b = *(const v2f*)(&Wt[(nt * 16 + l) * WTP + kk + 2 * half]);
      acc0[nt] = wmma4(a0, b, acc0[nt]);
      acc1[nt] = wmma4(a1, b, acc1[nt]);
    }
  }

#pragma unroll
  for (int nt = 0; nt < 4; ++nt) {
    const float bias = bl[k * C + nt * 16 + l];
#pragma unroll
    for (int j = 0; j < 8; ++j) {
      {
        const int row = r0 + j + 8 * half;
        const int b2 = row >> tshift, t2 = row & (T - 1);
        float v = acc0[nt][j] + bias;
        v = v > 0.f ? v : 0.f;
        Vout[((size_t)b2 * 256 + (size_t)k * T + t2) * C + nt * 16 + l] = v;
      }
      {
        const int row = r0 + 16 + j + 8 * half;
        const int b2 = row >> tshift, t2 = row & (T - 1);
        float v = acc1[nt][j] + bias;
        v = v > 0.f ? v : 0.f;
        Vout[((size_t)b2 * 256 + (size_t)k * T + t2) * C + nt * 16 + l] = v;
      }
    }
  }
}

// ---------------- Output layer: out[b][k*16+f] = V8[b][k][:] @ Fea[k]
__global__ __launch_bounds__(256) void butterfly_out_kernel(
    const float* __restrict__ V, const float* __restrict__ Fea,
    float* __restrict__ out) {
  const int tid  = threadIdx.x;
  const int wave = tid >> 5, lane = tid & 31;
  const int half = lane >> 4, l = lane & 15;
  const int k    = blockIdx.y;
  const int b0   = (blockIdx.x * 8 + wave) * 16;
  const float* arow = V + ((size_t)(b0 + l) * 256 + k) * C;
  const float* Fk   = Fea + (size_t)k * C * FOUT;

  v8f acc = {};
#pragma unroll
  for (int kk = 0; kk < C; kk += 4) {
    const v2f a = *(const v2f*)(arow + kk + 2 * half);
    v2f b;
    b.x = Fk[(kk + 2 * half + 0) * FOUT + l];
    b.y = Fk[(kk + 2 * half + 1) * FOUT + l];
    acc = wmma4(a, b, acc);
  }
#pragma unroll
  for (int j = 0; j < 8; ++j) {
    const int bb = b0 + j + 8 * half;
    out[(size_t)bb * (KMAX * FOUT) + k * FOUT + l] = acc[j];
  }
}

extern "C" void kernel_launch(void* const* d_in, const int* in_sizes, int n_in,
                              void* d_out, int out_size, void* d_ws,
                              size_t ws_size, hipStream_t stream) {
  (void)in_sizes; (void)n_in; (void)out_size; (void)ws_size;
  const float* in_data = (const float*)d_in[0];
  const float* W_in    = (const float*)d_in[1];
  const float* b_in    = (const float*)d_in[2];
  const float* W_lvl   = (const float*)d_in[3];
  const float* b_lvl   = (const float*)d_in[4];
  const float* Fea     = (const float*)d_in[5];
  float* out  = (float*)d_out;
  float* buf0 = (float*)d_ws;                       // 512*256*64 floats
  float* buf1 = buf0 + (size_t)B * KMAX * C;        // ping-pong partner

  // Input layer: 131072 rows / 128 rows-per-block = 1024 blocks.
  butterfly_input_kernel<<<dim3(ROWS / 128), 256, 0, stream>>>(in_data, W_in,
                                                               b_in, buf0);

  float* vin = buf0;
  float* vout = buf1;
  for (int lvl = 1; lvl <= NLVL; ++lvl) {
    const int K = 1 << lvl;
    const int tshift = NLVL - lvl;                  // T = 1 << tshift
    const int T = 1 << tshift;
    const int off = K - 2;
    dim3 grid(2 * T, K);                            // (B*T)/256 blocks per node
    const size_t shmem = (size_t)C * WTP * sizeof(float);  // 64*130*4 = 33.3 KB
    butterfly_level_kernel<<<grid, 256, shmem, stream>>>(
        vin, vout, W_lvl + (size_t)off * 2 * C * C, b_lvl + (size_t)off * C,
        tshift);
    float* t = vin; vin = vout; vout = t;
  }

  // After 8 levels V lives in buf0 again (even number of swaps).
  butterfly_out_kernel<<<dim3(B / 128, KMAX), 256, 0, stream>>>(vin, Fea, out);
}